// LRU_40303973106060
// MI455X (gfx1250) — compile-verified
//
#include <hip/hip_runtime.h>

typedef __attribute__((ext_vector_type(2))) float v2f;
typedef __attribute__((ext_vector_type(8))) float v8f;

#define BATCH   8
#define TLEN    4096
#define IDIM    512
#define HDIM    256
#define MROWS   (BATCH * TLEN)     // 32768
#define NCOLS   512                // N for both GEMMs (2H == I == 512)
#define KDIM    512                // K for both GEMMs
#define NCHUNK  32
#define CHUNKLEN (TLEN / NCHUNK)   // 128

// ---------------------------------------------------------------------------
// Kernel 1: per-channel recurrence coefficients
//   a = exp(-exp(nu)) * (cos(exp(theta)) + i sin(exp(theta)))
//   b = sqrt(1 - exp(-2 exp(nu)))
// ---------------------------------------------------------------------------
__global__ __launch_bounds__(256)
void lru_coeff(const float* __restrict__ nul, const float* __restrict__ thl,
               float* __restrict__ ar, float* __restrict__ ai,
               float* __restrict__ bv) {
    int h = threadIdx.x;
    float nu  = expf(nul[h]);
    float th  = expf(thl[h]);
    float mag = expf(-nu);
    ar[h] = mag * cosf(th);
    ai[h] = mag * sinf(th);
    bv[h] = sqrtf(1.0f - expf(-2.0f * nu));
}

// ---------------------------------------------------------------------------
// Kernel 2: Bcat[k, 0:256]   = B_re[k,h] * b[h]
//           Bcat[k, 256:512] = B_im[k,h] * b[h]
// ---------------------------------------------------------------------------
__global__ __launch_bounds__(256)
void lru_scaleB(const float* __restrict__ Bre, const float* __restrict__ Bim,
                const float* __restrict__ bv, float* __restrict__ Bcat) {
    int gid = blockIdx.x * 256 + threadIdx.x;   // [0, 512*256)
    int k = gid >> 8;
    int h = gid & 255;
    float s = bv[h];
    Bcat[(size_t)k * NCOLS + h]        = Bre[k * HDIM + h] * s;
    Bcat[(size_t)k * NCOLS + HDIM + h] = Bim[k * HDIM + h] * s;
}

// ---------------------------------------------------------------------------
// WMMA f32 GEMM: Out[M,512] = A[M,512] @ Bm[512,512]  (+ Xin*Dv epilogue)
// Block = 256 threads = 8 waves; block tile 256(M) x 64(N); wave tile 32x64.
// Per K-step of 4: 2 A-fragment b64 loads + 8 B-fragment b32 loads feed
// 8 V_WMMA_F32_16X16X4_F32 (each B fragment reused by two WMMAs).
// Fragment layouts per CDNA5 ISA 7.12.2:
//   A 16x4 : lanes 0-15 -> M=lane,   {K=k0,K=k0+1} in v[0],v[1]
//            lanes16-31 -> M=lane-16,{K=k0+2,k0+3}
//   B 4x16 : v[0]: K=k0 (lanes 0-15) / K=k0+2 (lanes 16-31), N=lane&15
//            v[1]: K=k0+1 / K=k0+3
//   C/D    : VGPR v -> row v (lanes 0-15) / row v+8 (lanes 16-31), N=lane&15
// ---------------------------------------------------------------------------
__global__ __launch_bounds__(256)
void lru_gemm_wmma(const float* __restrict__ A, const float* __restrict__ Bm,
                   float* __restrict__ Out,
                   const float* __restrict__ Xin, const float* __restrict__ Dv) {
    const int lane    = threadIdx.x & 31;
    const int wave    = threadIdx.x >> 5;
    const int m0      = blockIdx.y * 256 + wave * 32;
    const int n0      = blockIdx.x * 64;
    const int halfsel = lane >> 4;      // 0 for lanes 0-15, 1 for lanes 16-31
    const int lrow    = lane & 15;

    // acc[2*j + r] : N-tile j (j*16), M-subtile r (r*16)
    v8f acc[8];
#pragma unroll
    for (int i = 0; i < 8; ++i) acc[i] = (v8f){};

    const float* Arow0 = A  + (size_t)(m0 + lrow) * KDIM + (halfsel << 1);
    const float* Arow1 = Arow0 + (size_t)16 * KDIM;
    const float* Bcol  = Bm + (size_t)(halfsel << 1) * NCOLS + n0 + lrow;

    for (int k0 = 0; k0 < KDIM; k0 += 4) {
        v2f af0 = *(const v2f*)(Arow0 + k0);             // rows m0+lrow
        v2f af1 = *(const v2f*)(Arow1 + k0);             // rows m0+16+lrow
        // prefetch the streaming A operand ~1KB ahead
        __builtin_prefetch(Arow0 + k0 + 256, 0, 1);
        __builtin_prefetch(Arow1 + k0 + 256, 0, 1);
        const float* bp = Bcol + (size_t)k0 * NCOLS;
        v2f b0, b1, b2, b3;
        b0.x = bp[0];  b0.y = bp[NCOLS];
        b1.x = bp[16]; b1.y = bp[NCOLS + 16];
        b2.x = bp[32]; b2.y = bp[NCOLS + 32];
        b3.x = bp[48]; b3.y = bp[NCOLS + 48];
        acc[0] = __builtin_amdgcn_wmma_f32_16x16x4_f32(false, af0, false, b0, (short)0, acc[0], false, false);
        acc[1] = __builtin_amdgcn_wmma_f32_16x16x4_f32(false, af1, false, b0, (short)0, acc[1], false, false);
        acc[2] = __builtin_amdgcn_wmma_f32_16x16x4_f32(false, af0, false, b1, (short)0, acc[2], false, false);
        acc[3] = __builtin_amdgcn_wmma_f32_16x16x4_f32(false, af1, false, b1, (short)0, acc[3], false, false);
        acc[4] = __builtin_amdgcn_wmma_f32_16x16x4_f32(false, af0, false, b2, (short)0, acc[4], false, false);
        acc[5] = __builtin_amdgcn_wmma_f32_16x16x4_f32(false, af1, false, b2, (short)0, acc[5], false, false);
        acc[6] = __builtin_amdgcn_wmma_f32_16x16x4_f32(false, af0, false, b3, (short)0, acc[6], false, false);
        acc[7] = __builtin_amdgcn_wmma_f32_16x16x4_f32(false, af1, false, b3, (short)0, acc[7], false, false);
    }

#pragma unroll
    for (int r = 0; r < 2; ++r) {
        const int rbase = m0 + r * 16 + (halfsel << 3);
#pragma unroll
        for (int j = 0; j < 4; ++j) {
            int col = n0 + j * 16 + lrow;
            float dscale = Dv ? Dv[col] : 0.0f;
            v8f a = acc[2 * j + r];
#pragma unroll
            for (int v = 0; v < 8; ++v) {
                size_t idx = (size_t)(rbase + v) * NCOLS + col;
                float val = a[v];
                if (Xin) val = fmaf(Xin[idx], dscale, val);
                Out[idx] = val;
            }
        }
    }
}

// ---------------------------------------------------------------------------
// Scan pass 1: per-(b,h,chunk) local scan from zero state -> chunk carry
//   z_t = a*z_{t-1} + u_t over 128 steps
// ---------------------------------------------------------------------------
__global__ __launch_bounds__(256)
void lru_scan_pass1(const float* __restrict__ U, const float* __restrict__ ar,
                    const float* __restrict__ ai,
                    float* __restrict__ c_re, float* __restrict__ c_im) {
    int b = blockIdx.x >> 5;
    int c = blockIdx.x & 31;
    int h = threadIdx.x;
    float Ar = ar[h], Am = ai[h];
    float zr = 0.f, zm = 0.f;
    const float* up = U + (size_t)(b * TLEN + c * CHUNKLEN) * NCOLS + h;
    for (int i = 0; i < CHUNKLEN; ++i) {
        float ur = up[0];
        float ui = up[HDIM];
        float nr = fmaf(Ar, zr, fmaf(-Am, zm, ur));
        float nm = fmaf(Ar, zm, fmaf( Am, zr, ui));
        zr = nr; zm = nm;
        up += NCOLS;
    }
    int idx = (b * HDIM + h) * NCHUNK + c;
    c_re[idx] = zr;
    c_im[idx] = zm;
}

// ---------------------------------------------------------------------------
// Scan prefix: combine chunk carries per channel with a^128; record chunk
// initial states; write final state z[:, -1, :] (complex64 interleaved).
// ---------------------------------------------------------------------------
__global__ __launch_bounds__(256)
void lru_scan_prefix(const float* __restrict__ c_re, const float* __restrict__ c_im,
                     const float* __restrict__ ar, const float* __restrict__ ai,
                     const float* __restrict__ hxr, const float* __restrict__ hxi,
                     float* __restrict__ i_re, float* __restrict__ i_im,
                     float* __restrict__ out_state) {
    int b = blockIdx.x;
    int h = threadIdx.x;
    float pr = hxr[b * HDIM + h], pm = hxi[b * HDIM + h];
    // a^128 via 7 complex squarings
    float qr = ar[h], qm = ai[h];
#pragma unroll
    for (int s = 0; s < 7; ++s) {
        float nr = qr * qr - qm * qm;
        float nm = 2.0f * qr * qm;
        qr = nr; qm = nm;
    }
    int base = (b * HDIM + h) * NCHUNK;
    for (int c = 0; c < NCHUNK; ++c) {
        i_re[base + c] = pr;
        i_im[base + c] = pm;
        float cr = c_re[base + c], cm = c_im[base + c];
        float nr = fmaf(qr, pr, fmaf(-qm, pm, cr));
        float nm = fmaf(qr, pm, fmaf( qm, pr, cm));
        pr = nr; pm = nm;
    }
    out_state[(b * HDIM + h) * 2]     = pr;
    out_state[(b * HDIM + h) * 2 + 1] = pm;
}

// ---------------------------------------------------------------------------
// Scan pass 2: re-scan each chunk from its correct initial state, write
// Z2[b,t,2h]=re, Z2[b,t,2h+1]=im (torch view_as_real interleave) as one
// 8-byte store per step.
// ---------------------------------------------------------------------------
__global__ __launch_bounds__(256)
void lru_scan_pass2(const float* __restrict__ U, const float* __restrict__ ar,
                    const float* __restrict__ ai,
                    const float* __restrict__ i_re, const float* __restrict__ i_im,
                    float* __restrict__ Z2) {
    int b = blockIdx.x >> 5;
    int c = blockIdx.x & 31;
    int h = threadIdx.x;
    float Ar = ar[h], Am = ai[h];
    int base = (b * HDIM + h) * NCHUNK + c;
    float zr = i_re[base], zm = i_im[base];
    size_t row = (size_t)(b * TLEN + c * CHUNKLEN);
    const float* up = U  + row * NCOLS + h;
    float*       zp = Z2 + row * NCOLS + 2 * h;      // 8-byte aligned
    for (int i = 0; i < CHUNKLEN; ++i) {
        float ur = up[0];
        float ui = up[HDIM];
        float nr = fmaf(Ar, zr, fmaf(-Am, zm, ur));
        float nm = fmaf(Ar, zm, fmaf( Am, zr, ui));
        zr = nr; zm = nm;
        v2f z; z.x = zr; z.y = zm;
        *(v2f*)zp = z;                               // single b64 store
        up += NCOLS;
        zp += NCOLS;
    }
}

// ---------------------------------------------------------------------------
extern "C" void kernel_launch(void* const* d_in, const int* in_sizes, int n_in,
                              void* d_out, int out_size, void* d_ws, size_t ws_size,
                              hipStream_t stream) {
    const float* x   = (const float*)d_in[0];   // [8,4096,512]
    const float* hxr = (const float*)d_in[1];   // [8,256]
    const float* hxi = (const float*)d_in[2];   // [8,256]
    const float* nul = (const float*)d_in[3];   // [256]
    const float* thl = (const float*)d_in[4];   // [256]
    const float* Bre = (const float*)d_in[5];   // [512,256]
    const float* Bim = (const float*)d_in[6];   // [512,256]
    const float* Cm  = (const float*)d_in[7];   // [512,512]
    const float* Dv  = (const float*)d_in[8];   // [512]

    float* y         = (float*)d_out;                  // [8,4096,512]
    float* state_out = y + (size_t)MROWS * NCOLS;      // [8,256] complex64 interleaved

    // Workspace layout (floats)
    float* ws   = (float*)d_ws;
    float* a_re = ws;                                   // 256
    float* a_im = a_re + 256;                           // 256
    float* bvec = a_im + 256;                           // 256 (+pad)
    float* Bcat = ws + 1024;                            // 512*512
    float* U    = Bcat + (size_t)KDIM * NCOLS;          // 32768*512
    float* Z2   = U    + (size_t)MROWS * NCOLS;         // 32768*512
    float* c_re = Z2   + (size_t)MROWS * NCOLS;         // 8*256*32
    float* c_im = c_re + BATCH * HDIM * NCHUNK;
    float* i_re = c_im + BATCH * HDIM * NCHUNK;
    float* i_im = i_re + BATCH * HDIM * NCHUNK;

    lru_coeff<<<1, 256, 0, stream>>>(nul, thl, a_re, a_im, bvec);
    lru_scaleB<<<(IDIM * HDIM) / 256, 256, 0, stream>>>(Bre, Bim, bvec, Bcat);

    dim3 gg(NCOLS / 64, MROWS / 256);
    lru_gemm_wmma<<<gg, 256, 0, stream>>>(x, Bcat, U, nullptr, nullptr);

    lru_scan_pass1<<<BATCH * NCHUNK, 256, 0, stream>>>(U, a_re, a_im, c_re, c_im);
    lru_scan_prefix<<<BATCH, 256, 0, stream>>>(c_re, c_im, a_re, a_im, hxr, hxi,
                                               i_re, i_im, state_out);
    lru_scan_pass2<<<BATCH * NCHUNK, 256, 0, stream>>>(U, a_re, a_im, i_re, i_im, Z2);

    lru_gemm_wmma<<<gg, 256, 0, stream>>>(Z2, Cm, y, x, Dv);
}